// ReacFullGbModel_21466246546026
// MI455X (gfx1250) — compile-verified
//
#include <hip/hip_runtime.h>

// ---------------------------------------------------------------------------
// ReacFullGbModel on gfx1250 (MI455X), compile-only tuning.
//
// Mapping: one wave32 owns 16 batch elements (the N columns of WMMA tiles).
// Lanes l and l^16 hold mirrored per-batch scalar state for batch (base+l%16),
// so both lane halves can supply the K-rows of WMMA B operands with a single
// select (no shuffles). Every tiny matvec (1x16 and 12x32) runs as
// V_WMMA_F32_16X16X4_F32 with hidden on M / batch on N; the 16->1 (32->1)
// output layers reduce over M = in-register over the 8 accumulator VGPRs
// plus one SWAPX16 ds_swizzle.  Serial scan over T=1024 stays in registers.
// ---------------------------------------------------------------------------

#define BATCH  2048
#define TSTEPS 1024

typedef __attribute__((ext_vector_type(2))) float v2f;
typedef __attribute__((ext_vector_type(8))) float v8f;

__device__ __forceinline__ v8f wmma4(v2f a, v2f b, v8f c) {
  // D = A(16x4) x B(4x16) + C(16x16), fp32
  return __builtin_amdgcn_wmma_f32_16x16x4_f32(false, a, false, b, (short)0, c,
                                               false, false);
}

__device__ __forceinline__ float fast_tanh(float x) {
#if __has_builtin(__builtin_amdgcn_tanhf)
  return __builtin_amdgcn_tanhf(x);
#elif __has_builtin(__builtin_amdgcn_tanh_f32)
  return __builtin_amdgcn_tanh_f32(x);
#else
  // tanh(x) = 1 - 2/(exp2(2*log2e*x)+1)  -> v_exp_f32 + v_rcp_f32 (TRANS ops)
  float e = __builtin_amdgcn_exp2f(2.8853900817779268f * x);
  return 1.0f - 2.0f * __builtin_amdgcn_rcpf(e + 1.0f);
#endif
}

// exchange value with lane^16 (SWAPX16 group-of-32 swizzle: xor=0x10,and=0x1f)
__device__ __forceinline__ float swap16(float x) {
  return __int_as_float(__builtin_amdgcn_ds_swizzle(__float_as_int(x), 0x401f));
}

struct RNet {   // 1 -> 16 tanh -> 1
  v2f  A;       // W0^T as 16x4 A-operand (only K=0 column nonzero)
  v8f  b0;      // bias, per-lane view of C layout: c[v] = b0[v + 8*(lane>=16)]
  v8f  w1;      // output weights, same per-lane view
  float b1;
};

__device__ __forceinline__ RNet load_rnet(const float* W0, const float* b0,
                                          const float* W1, const float* b1,
                                          int m, bool lo, int hoff) {
  RNet w;
  w.A.x = lo ? W0[m] : 0.0f;   // lanes16-31 carry K=2 row -> zero
  w.A.y = 0.0f;                // K=1 / K=3 rows -> zero
#pragma unroll
  for (int v = 0; v < 8; ++v) {
    w.b0[v] = b0[hoff + v];
    w.w1[v] = W1[hoff + v];
  }
  w.b1 = b1[0];
  return w;
}

// r = W1^T tanh(W0^T * s + b0) + b1 for 16 batch columns at once
__device__ __forceinline__ float rnet_eval(const RNet& w, float s, bool lo) {
  v2f B;
  B.x = lo ? s : 0.0f;   // K=0 row = input scalar per batch column
  B.y = 0.0f;
  v8f d = wmma4(w.A, B, w.b0);
  float p = 0.0f;
#pragma unroll
  for (int v = 0; v < 8; ++v) p += fast_tanh(d[v]) * w.w1[v];
  return (p + swap16(p)) + w.b1;   // identical in both lane halves
}

struct Der { float a, b, c; };

__device__ __forceinline__ Der fxu(float Ca, float Cb, float Cc, float Caf,
                                   const RNet& n1, const RNet& n2,
                                   const RNet& n3, bool lo) {
  const float CASTD = 0.5f, CBSTD = 0.3f, CAMEAN = 1.0f, CBMEAN = 0.5f;
  const float Fv = 0.1f, Vinv = 1.0f;
  float r1 = rnet_eval(n1, Ca, lo) * CASTD;
  float r2 = rnet_eval(n2, Cb, lo) * CBSTD;
  float r3 = rnet_eval(n3, Cc, lo) * CBSTD;
  float CaP = Ca * CASTD + CAMEAN;
  float CbP = Cb * CBSTD + CBMEAN;
  float CcP = Cc * CBSTD + CBMEAN;
  float dCa = Fv * (Caf - CaP) * Vinv - r1;
  float dCb = -Fv * CbP * Vinv + r1 - 3.0f * r2 + r3;
  float dCc = -Fv * CcP * Vinv + r2 - r3;
  Der d;
  d.a = dCa * (1.0f / CASTD);
  d.b = dCb * (1.0f / CBSTD);
  d.c = dCc * (1.0f / CBSTD);
  return d;
}

__global__ __launch_bounds__(32) void ReacFullGbModel_kernel(
    const float* __restrict__ useq, const float* __restrict__ x0,
    const float* __restrict__ r1W0, const float* __restrict__ r1b0,
    const float* __restrict__ r1W1, const float* __restrict__ r1b1,
    const float* __restrict__ r2W0, const float* __restrict__ r2b0,
    const float* __restrict__ r2W1, const float* __restrict__ r2b1,
    const float* __restrict__ r3W0, const float* __restrict__ r3b0,
    const float* __restrict__ r3W1, const float* __restrict__ r3b1,
    const float* __restrict__ eW0, const float* __restrict__ eb0,
    const float* __restrict__ eW1, const float* __restrict__ eb1,
    float* __restrict__ out) {
  const int lane = threadIdx.x;          // one wave per block
  const bool lo  = lane < 16;
  const int  m   = lane & 15;
  const int  hoff = lo ? 0 : 8;          // which 8 hidden rows this lane sees
  const int  b   = blockIdx.x * 16 + m;  // mirrored batch element

  // ---- load reaction-net weights (constant over T) ----
  RNet n1 = load_rnet(r1W0, r1b0, r1W1, r1b1, m, lo, hoff);
  RNet n2 = load_rnet(r2W0, r2b0, r2W1, r2b1, m, lo, hoff);
  RNet n3 = load_rnet(r3W0, r3b0, r3W1, r3b1, m, lo, hoff);

  // ---- estC: 12 -> 32 tanh -> 1, as 2 hidden tiles x 3 K=4 chunks ----
  v2f Ae[2][3];
  v8f be[2], w1e[2];
#pragma unroll
  for (int t = 0; t < 2; ++t) {
    const int h = m + 16 * t;            // A rows live in lane%16
#pragma unroll
    for (int c = 0; c < 3; ++c) {
      const int k0 = 4 * c + (lo ? 0 : 2);   // VGPR0 = K0/K2, VGPR1 = K1/K3
      Ae[t][c].x = eW0[k0 * 32 + h];
      Ae[t][c].y = eW0[(k0 + 1) * 32 + h];
    }
#pragma unroll
    for (int v = 0; v < 8; ++v) {
      be[t][v]  = eb0[16 * t + hoff + v];
      w1e[t][v] = eW1[16 * t + hoff + v];
    }
  }
  const float b1e = eb1[0];

  // ---- initial state (mirrored in both lane halves) ----
  const float* xp = x0 + (size_t)b * 15;
  float xa = xp[0], xb = xp[1], xc = xp[2];
  float z[12];
#pragma unroll
  for (int j = 0; j < 12; ++j) z[j] = xp[3 + j];

  const float* ub = useq + (size_t)b * TSTEPS;
  float* ob = out + (size_t)b * TSTEPS * 4;

#pragma unroll 1
  for (int t = 0; t < TSTEPS; ++t) {
    float u = ub[t];
    if (t + 64 < TSTEPS) __builtin_prefetch(&ub[t + 64], 0, 0);

    // ---- CcNN = estC(z): 6 chained WMMAs, bias pre-loaded into C ----
    v8f d0 = be[0], d1 = be[1];
#pragma unroll
    for (int c = 0; c < 3; ++c) {
      v2f Bz;
      Bz.x = lo ? z[4 * c + 0] : z[4 * c + 2];
      Bz.y = lo ? z[4 * c + 1] : z[4 * c + 3];
      d0 = wmma4(Ae[0][c], Bz, d0);
      d1 = wmma4(Ae[1][c], Bz, d1);
    }
    float p = 0.0f;
#pragma unroll
    for (int v = 0; v < 8; ++v)
      p += fast_tanh(d0[v]) * w1e[0][v] + fast_tanh(d1[v]) * w1e[1][v];
    float CcNN = (p + swap16(p)) + b1e;

    // ---- emit y = [x, CcNN] (pre-update state), lanes 0-15 only ----
    if (lo) {
      float4 y;
      y.x = xa; y.y = xb; y.z = xc; y.w = CcNN;
      *reinterpret_cast<float4*>(ob + (size_t)t * 4) = y;
    }

    // ---- RK4 on the 3-state ODE (DELTA = 1) ----
    const float Caf = u * 0.5f + 2.0f;   // u*CAFSTD + CAFMEAN
    Der k1 = fxu(xa, xb, xc, Caf, n1, n2, n3, lo);
    Der k2 = fxu(xa + 0.5f * k1.a, xb + 0.5f * k1.b, xc + 0.5f * k1.c, Caf,
                 n1, n2, n3, lo);
    Der k3 = fxu(xa + 0.5f * k2.a, xb + 0.5f * k2.b, xc + 0.5f * k2.c, Caf,
                 n1, n2, n3, lo);
    Der k4 = fxu(xa + k3.a, xb + k3.b, xc + k3.c, Caf, n1, n2, n3, lo);
    const float s6 = 1.0f / 6.0f;
    float nxa = xa + s6 * (k1.a + 2.0f * (k2.a + k3.a) + k4.a);
    float nxb = xb + s6 * (k1.b + 2.0f * (k2.b + k3.b) + k4.b);
    float nxc = xc + s6 * (k1.c + 2.0f * (k2.c + k3.c) + k4.c);

    // ---- z <- [ypseq[2:], x[:2], upseq[1:], u] using PRE-update x ----
    z[0] = z[2]; z[1] = z[3]; z[2] = z[4]; z[3] = z[5];
    z[4] = z[6]; z[5] = z[7];
    z[6] = xa;   z[7] = xb;
    z[8] = z[9]; z[9] = z[10]; z[10] = z[11]; z[11] = u;

    xa = nxa; xb = nxb; xc = nxc;
  }
}

extern "C" void kernel_launch(void* const* d_in, const int* in_sizes, int n_in,
                              void* d_out, int out_size, void* d_ws,
                              size_t ws_size, hipStream_t stream) {
  (void)in_sizes; (void)n_in; (void)out_size; (void)d_ws; (void)ws_size;
  const float* p[18];
  for (int i = 0; i < 18; ++i) p[i] = (const float*)d_in[i];
  dim3 grid(BATCH / 16), block(32);   // 128 independent waves
  ReacFullGbModel_kernel<<<grid, block, 0, stream>>>(
      p[0], p[1], p[2], p[3], p[4], p[5], p[6], p[7], p[8], p[9], p[10], p[11],
      p[12], p[13], p[14], p[15], p[16], p[17], (float*)d_out);
}